// development_model_10746008175151
// MI455X (gfx1250) — compile-verified
//
#include <hip/hip_runtime.h>

// DevNet path-signature model for MI455X (gfx1250, wave32, WMMA):
//   per chain n: B_t = sum_i dX[t,i] * G_i (32x32 skew), M_t = expm(B_t),
//   Z_n = prod_t M_t, out = fc(Z_n.flatten()).
// One 4-wave workgroup per chain; all 32x32 fp32 matmuls on
// V_WMMA_F32_16X16X4_F32 with two independent accumulation chains per wave.
// expm scaling exponent is derived from the precomputed generator norms so
// every thread computes it locally (no reduction, no broadcast barrier).

typedef float v2f __attribute__((ext_vector_type(2)));
typedef float v8f __attribute__((ext_vector_type(8)));

#define Nn   128
#define Tt   256
#define Ii   8
#define Mm   32
#define OUTn 10

// dst = a @ b (32x32 fp32). Wave w computes tile (w>>1, w&1).
// Two independent WMMA chains (K = 0,8,16,24 and K = 4,12,20,28) for ILP.
// If FUSE: epilogue writes dst = D*ik and accumulates E += D*ik (each output
// element is owned by exactly one lane -> race-free).
template <bool FUSE>
__device__ __forceinline__ void mm32_t(float* __restrict__ dst,
                                       const float* __restrict__ a,
                                       const float* __restrict__ b,
                                       float* __restrict__ eacc, float ik,
                                       int wave, int lane) {
    __syncthreads();                        // operands ready
    const int mi = (wave >> 1) & 1;
    const int nj = wave & 1;
    const int lm = lane & 15;
    const int hi = lane >> 4;               // 0: lanes 0-15, 1: lanes 16-31
    v8f c0 = {0.f, 0.f, 0.f, 0.f, 0.f, 0.f, 0.f, 0.f};
    v8f c1 = {0.f, 0.f, 0.f, 0.f, 0.f, 0.f, 0.f, 0.f};
#pragma unroll
    for (int k0 = 0; k0 < Mm; k0 += 8) {
        v2f av0, bv0, av1, bv1;
        const int ar = (mi * 16 + lm) * Mm + 2 * hi;
        av0.x = a[ar + k0 + 0];
        av0.y = a[ar + k0 + 1];
        av1.x = a[ar + k0 + 4];
        av1.y = a[ar + k0 + 5];
        const int br = (2 * hi) * Mm + nj * 16 + lm;
        bv0.x = b[br + (k0 + 0) * Mm];
        bv0.y = b[br + (k0 + 1) * Mm];
        bv1.x = b[br + (k0 + 4) * Mm];
        bv1.y = b[br + (k0 + 5) * Mm];
        c0 = __builtin_amdgcn_wmma_f32_16x16x4_f32(false, av0, false, bv0,
                                                   (short)0, c0, false, false);
        c1 = __builtin_amdgcn_wmma_f32_16x16x4_f32(false, av1, false, bv1,
                                                   (short)0, c1, false, false);
    }
    const v8f c = c0 + c1;
#pragma unroll
    for (int v = 0; v < 8; ++v) {
        const int idx = (mi * 16 + v + 8 * hi) * Mm + nj * 16 + lm;
        float d = c[v];
        if (FUSE) {
            d *= ik;
            eacc[idx] += d;
        }
        dst[idx] = d;
    }
    __syncthreads();                        // dst (and E) visible to all waves
}

__device__ __forceinline__ void mm32(float* dst, const float* a, const float* b,
                                     int wave, int lane) {
    mm32_t<false>(dst, a, b, nullptr, 0.f, wave, lane);
}

__global__ __launch_bounds__(128)
void devnet_chain_kernel(const float* __restrict__ X,
                         const float* __restrict__ A,
                         const float* __restrict__ fcw,
                         const float* __restrict__ fcb,
                         float* __restrict__ out) {
    __shared__ float sG[Ii * Mm * Mm];      // 8 skew generators, 32 KB
    __shared__ float mats[6 * Mm * Mm];     // P, E, Ta, Tb, Za, Zb, 24 KB
    __shared__ float sdx[Ii];
    __shared__ float sred[2 * 128];         // row-norm scratch / FC scratch
    __shared__ float sgn[Ii];               // ||G_i||_inf

    const int n    = blockIdx.x;
    const int tid  = threadIdx.x;
    const int lane = tid & 31;
    const int wave = tid >> 5;

    float* P  = mats;
    float* E  = mats + 1024;
    float* Ta = mats + 2048;
    float* Tb = mats + 3072;
    float* Za = mats + 4096;
    float* Zb = mats + 5120;

    // --- Build skew generators: G_i = tril(A^T_i,-1) - tril(A^T_i,-1)^T ---
    for (int j = tid; j < Ii * 1024; j += 128) {
        const int i = j >> 10, pq = j & 1023, p = pq >> 5, q = pq & 31;
        float g = 0.f;
        if (p > q)      g =  A[(p * Mm + q) * Ii + i];
        else if (p < q) g = -A[(q * Mm + p) * Ii + i];
        sG[j] = g;
    }
    // Z = I
    for (int j = tid; j < 1024; j += 128)
        Za[j] = ((j >> 5) == (j & 31)) ? 1.f : 0.f;
    __syncthreads();

    // --- Generator inf-norms (once per block): 256 rows, 2 per thread ---
    {
        float rs0 = 0.f, rs1 = 0.f;
        for (int q = 0; q < Mm; ++q) {
            rs0 += fabsf(sG[tid * Mm + q]);
            rs1 += fabsf(sG[(tid + 128) * Mm + q]);
        }
        sred[tid]       = rs0;
        sred[tid + 128] = rs1;
    }
    __syncthreads();
    if (tid < Ii) {
        float mx = 0.f;
        for (int r = 0; r < Mm; ++r) mx = fmaxf(mx, sred[tid * Mm + r]);
        sgn[tid] = mx;
    }
    __syncthreads();

    float gnr[Ii];
#pragma unroll
    for (int i = 0; i < Ii; ++i) gnr[i] = sgn[i];

    const float* Xr = X + (size_t)n * (Tt * Ii);

    for (int t = 0; t < Tt - 1; ++t) {
        if (t + 2 < Tt) __builtin_prefetch(Xr + (t + 2) * Ii, 0, 1);
        if (tid < Ii) sdx[tid] = Xr[(t + 1) * Ii + tid] - Xr[t * Ii + tid];
        __syncthreads();

        float dx[Ii];
#pragma unroll
        for (int i = 0; i < Ii; ++i) dx[i] = sdx[i];

        // Scaling exponent from a-priori bound ||B||inf <= sum|dx_i|*||G_i||inf.
        // Identical in every thread -> no reduction, no broadcast.
        float mx = 0.f;
#pragma unroll
        for (int i = 0; i < Ii; ++i) mx = fmaf(fabsf(dx[i]), gnr[i], mx);
        int s = 0;
        while (mx > 0.25f && s < 16) { mx *= 0.5f; ++s; }
        const float scale = __builtin_ldexpf(1.0f, -s);

        // One pass: P = (sum_i dx_i G_i)*2^-s ; Ta = P ; E = I + P
        for (int j = tid; j < 1024; j += 128) {
            float acc = 0.f;
#pragma unroll
            for (int i = 0; i < Ii; ++i) acc = fmaf(dx[i], sG[i * 1024 + j], acc);
            const float p = acc * scale;
            P[j]  = p;
            Ta[j] = p;
            E[j]  = p + (((j >> 5) == (j & 31)) ? 1.f : 0.f);
        }
        // (mm32 entry barrier covers visibility)

        // Taylor k=2..7 fused: Tb = (Ta@P)/k ; E += Tb   (remainder ~1e-9)
#pragma unroll 1
        for (int k = 2; k <= 7; ++k) {
            mm32_t<true>(Tb, Ta, P, E, 1.f / (float)k, wave, lane);
            float* tmp = Ta; Ta = Tb; Tb = tmp;
        }

        // Squaring: E <- E@E, s times (uniform trip count across the block)
        for (int sq = 0; sq < s; ++sq) {
            mm32(Ta, E, E, wave, lane);
            float* tmp = E; E = Ta; Ta = tmp;
        }

        // Z <- Z @ M_t
        mm32(Zb, Za, E, wave, lane);
        { float* tmp = Za; Za = Zb; Zb = tmp; }
    }

    // --- FC head: out[n,o] = fc_b[o] + <fc_w[o,:], Z.flatten()> ---
    const float* Z = Za;
    for (int o = 0; o < OUTn; ++o) {
        float part = 0.f;
        for (int k = tid; k < 1024; k += 128)
            part = fmaf(fcw[o * 1024 + k], Z[k], part);
        sred[tid] = part;
        __syncthreads();
        if (tid == 0) {
            float acc = fcb[o];
            for (int r = 0; r < 128; ++r) acc += sred[r];
            out[n * OUTn + o] = acc;
        }
        __syncthreads();
    }
}

extern "C" void kernel_launch(void* const* d_in, const int* in_sizes, int n_in,
                              void* d_out, int out_size, void* d_ws, size_t ws_size,
                              hipStream_t stream) {
    const float* X   = (const float*)d_in[0];   // (128, 256, 8)
    const float* A   = (const float*)d_in[1];   // (1, 32, 32, 8)
    const float* fcw = (const float*)d_in[2];   // (10, 1024)
    const float* fcb = (const float*)d_in[3];   // (10,)
    float* out = (float*)d_out;                 // (128, 10)
    (void)in_sizes; (void)n_in; (void)out_size; (void)d_ws; (void)ws_size;

    devnet_chain_kernel<<<dim3(Nn), dim3(128), 0, stream>>>(X, A, fcw, fcb, out);
}